// EquivariantLayerNorm_25022479466496
// MI455X (gfx1250) — compile-verified
//
#include <hip/hip_runtime.h>
#include <cstdint>

// EquivariantLayerNorm: N x 480 fp32 rows.
//   [0,128)   : LayerNorm(128) * weight + bias
//   [128,320) : 64 segments of 3, per-segment normalize
//   [320,480) : 32 segments of 5, per-segment normalize
//
// Memory-bound (≈1.0 GB total traffic → ~43us floor at 23.3 TB/s).
// Strategy: async-DMA each block's contiguous 15360B chunk global->LDS
// (b128, fully coalesced, no VGPR staging), normalize in place in LDS,
// async-DMA LDS->global. Wave32 shuffle allreduce for the 128-wide norm.

#define EPSV 1e-5f
#define SDIM 128
#define ROW_DIM 480
#define ROWS_PER_BLOCK 8
#define BLOCK_FLOATS (ROWS_PER_BLOCK * ROW_DIM)  // 3840
#define BLOCK_VEC4 (BLOCK_FLOATS / 4)            // 960

__global__ __launch_bounds__(256)
void eqln_kernel(const float* __restrict__ x,
                 const float* __restrict__ weight,
                 const float* __restrict__ bias,
                 float* __restrict__ out)
{
    __shared__ __align__(16) float smem[BLOCK_FLOATS];

    const int tid  = threadIdx.x;
    const int wave = tid >> 5;
    const int lane = tid & 31;

    // Low 32 bits of a generic pointer into LDS == wave-relative LDS byte
    // offset (flat->LDS aperture mapping truncates to addr[31:0]).
    const uint32_t lds_base = (uint32_t)(uintptr_t)&smem[0];

    const size_t blockBase = (size_t)blockIdx.x * BLOCK_FLOATS;
    const float* gsrc = x + blockBase;
    float*       gdst = out + blockBase;

    // ---------------- async stage-in: global -> LDS, b128 ----------------
    for (int k = tid; k < BLOCK_VEC4; k += 256) {
        uint32_t loff = lds_base + (uint32_t)(k * 16);
        uint64_t ga   = (uint64_t)(uintptr_t)(gsrc + k * 4);
        asm volatile("global_load_async_to_lds_b128 %0, %1, off"
                     :: "v"(loff), "v"(ga) : "memory");
    }
    asm volatile("s_wait_asynccnt 0" ::: "memory");
    __syncthreads();

    float* row = smem + wave * ROW_DIM;

    // ---------------- phase A: LayerNorm over first 128 ----------------
    {
        float4 v = ((float4*)row)[lane];
        float s  = v.x + v.y + v.z + v.w;
        float ss = v.x * v.x + v.y * v.y + v.z * v.z + v.w * v.w;
#pragma unroll
        for (int off = 16; off >= 1; off >>= 1) {
            s  += __shfl_xor(s,  off, 32);
            ss += __shfl_xor(ss, off, 32);
        }
        float mean = s * (1.0f / 128.0f);
        float var  = ss * (1.0f / 128.0f) - mean * mean;
        float rn   = rsqrtf(var + EPSV);

        float4 w4 = ((const float4*)weight)[lane];
        float4 b4 = ((const float4*)bias)[lane];
        float4 r;
        r.x = (v.x - mean) * rn * w4.x + b4.x;
        r.y = (v.y - mean) * rn * w4.y + b4.y;
        r.z = (v.z - mean) * rn * w4.z + b4.z;
        r.w = (v.w - mean) * rn * w4.w + b4.w;
        ((float4*)row)[lane] = r;
    }

    // ------------- phase B: 64 x (3-elem) segments, 2 per lane -------------
    {
        float* p = row + SDIM + 6 * lane;        // even index -> 8B aligned
        float2 q0 = *(float2*)(p);
        float2 q1 = *(float2*)(p + 2);
        float2 q2 = *(float2*)(p + 4);

        float m0 = (q0.x + q0.y + q1.x) * (1.0f / 3.0f);
        float d0 = q0.x - m0, d1 = q0.y - m0, d2 = q1.x - m0;
        float r0 = rsqrtf((d0 * d0 + d1 * d1 + d2 * d2) * (1.0f / 3.0f) + EPSV);

        float m1 = (q1.y + q2.x + q2.y) * (1.0f / 3.0f);
        float e0 = q1.y - m1, e1 = q2.x - m1, e2 = q2.y - m1;
        float r1 = rsqrtf((e0 * e0 + e1 * e1 + e2 * e2) * (1.0f / 3.0f) + EPSV);

        q0.x = d0 * r0; q0.y = d1 * r0; q1.x = d2 * r0;
        q1.y = e0 * r1; q2.x = e1 * r1; q2.y = e2 * r1;
        *(float2*)(p)     = q0;
        *(float2*)(p + 2) = q1;
        *(float2*)(p + 4) = q2;
    }

    // ------------- phase C: 32 x (5-elem) segments, 1 per lane -------------
    {
        float* p = row + SDIM + 192 + 5 * lane;  // stride 5 over 64 banks: conflict-free
        float a0 = p[0], a1 = p[1], a2 = p[2], a3 = p[3], a4 = p[4];
        float m  = (a0 + a1 + a2 + a3 + a4) * 0.2f;
        float d0 = a0 - m, d1 = a1 - m, d2 = a2 - m, d3 = a3 - m, d4 = a4 - m;
        float rv = rsqrtf((d0 * d0 + d1 * d1 + d2 * d2 + d3 * d3 + d4 * d4) * 0.2f + EPSV);
        p[0] = d0 * rv; p[1] = d1 * rv; p[2] = d2 * rv; p[3] = d3 * rv; p[4] = d4 * rv;
    }

    __syncthreads();

    // ---------------- async stage-out: LDS -> global, b128 ----------------
    for (int k = tid; k < BLOCK_VEC4; k += 256) {
        uint32_t loff = lds_base + (uint32_t)(k * 16);
        uint64_t ga   = (uint64_t)(uintptr_t)(gdst + k * 4);
        asm volatile("global_store_async_from_lds_b128 %0, %1, off"
                     :: "v"(ga), "v"(loff) : "memory");
    }
    asm volatile("s_wait_asynccnt 0" ::: "memory");
}

extern "C" void kernel_launch(void* const* d_in, const int* in_sizes, int n_in,
                              void* d_out, int out_size, void* d_ws, size_t ws_size,
                              hipStream_t stream) {
    const float* x      = (const float*)d_in[0];
    const float* weight = (const float*)d_in[1];
    const float* bias   = (const float*)d_in[2];
    float* out          = (float*)d_out;

    const int n_rows = in_sizes[0] / ROW_DIM;           // 262144
    const int blocks = n_rows / ROWS_PER_BLOCK;         // 32768

    hipLaunchKernelGGL(eqln_kernel, dim3(blocks), dim3(256), 0, stream,
                       x, weight, bias, out);
}